// Representation_23502061044385
// MI455X (gfx1250) — compile-verified
//
#include <hip/hip_runtime.h>
#include <hip/hip_bf16.h>
#include <math.h>

#define BS    128
#define NC    1024
#define U     512
#define FEAT  512
#define FOURU 2048

// LDS A-tile row stride in floats: 516 % 64 == 4 -> conflict-free frag reads
#define APAD  516

typedef __attribute__((ext_vector_type(2))) float v2f;
typedef __attribute__((ext_vector_type(8))) float v8f;

// ---------------------------------------------------------------------------
// CDNA5 async DMA: copy 16B/lane from global to LDS, tracked by ASYNCcnt.
// lds_off is the wave-relative LDS byte offset (low 32 bits of a generic
// pointer to a __shared__ object).
// ---------------------------------------------------------------------------
__device__ __forceinline__ void async_copy_b128(unsigned lds_off,
                                                const void* gptr) {
  asm volatile("global_load_async_to_lds_b128 %0, %1, off"
               :: "v"(lds_off), "v"(gptr)
               : "memory");
}

__device__ __forceinline__ void wait_asynccnt0() {
  asm volatile("s_wait_asynccnt 0x0" ::: "memory");
}

__device__ __forceinline__ unsigned lds_offset(const void* shared_ptr) {
  // Generic pointers to LDS carry the wave-relative offset in the low 32 bits
  // (flat->LDS mapping truncates addr[31:0]); that is what the async
  // instruction's LDS-address VGPR expects.
  return (unsigned)(unsigned long long)(uintptr_t)shared_ptr;
}

// ---------------------------------------------------------------------------
// Step 1: row-wise argmax of class_logits (softmax(logits*1e10) is exactly
// one-hot at the argmax in fp32), one block per batch row.
// ---------------------------------------------------------------------------
__global__ __launch_bounds__(256) void argmax_kernel(
    const float* __restrict__ logits, int* __restrict__ idx) {
  __shared__ float sv[256];
  __shared__ int   si[256];
  const int b = blockIdx.x;
  const float* row = logits + b * NC;
  float best = -INFINITY;
  int   bi   = 0;
  for (int c = threadIdx.x; c < NC; c += 256) {
    float v = row[c];
    if (v > best) { best = v; bi = c; }
  }
  sv[threadIdx.x] = best;
  si[threadIdx.x] = bi;
  __syncthreads();
  for (int s = 128; s > 0; s >>= 1) {
    if (threadIdx.x < s) {
      float ov = sv[threadIdx.x + s];
      int   oi = si[threadIdx.x + s];
      if (ov > sv[threadIdx.x] ||
          (ov == sv[threadIdx.x] && oi < si[threadIdx.x])) {
        sv[threadIdx.x] = ov;
        si[threadIdx.x] = oi;
      }
    }
    __syncthreads();
  }
  if (threadIdx.x == 0) idx[b] = si[0];
}

// ---------------------------------------------------------------------------
// Step 2: z = [x | h_states[idx]] @ [kernel ; rec_kernel]  (128x1024)@(1024x2048)
// fp32 WMMA 16x16x4. Per block: one 16-row M tile, 256 N columns (8 waves x
// 16x32 strips). The shared A tile (16x512 fp32, 33KB w/ pad) is staged into
// LDS once per phase via global_load_async_to_lds_b128 (ASYNCcnt), removing
// the 8x-redundant per-wave A global loads; waves read A frags with
// conflict-free ds_load_b64. B is read directly from global (disjoint columns
// per wave, L2-resident).
// ---------------------------------------------------------------------------
__global__ __launch_bounds__(256) void gemm_z_kernel(
    const float* __restrict__ x,         // (128,512)
    const float* __restrict__ h_states,  // (1024,512)
    const float* __restrict__ kernelW,   // (512,2048)
    const float* __restrict__ recW,      // (512,2048)
    const int*   __restrict__ idx,       // (128)
    float* __restrict__ z)               // (128,2048) raw (no bias)
{
  __shared__ float sA[16][APAD];

  const int tid   = threadIdx.x;
  const int lane  = tid & 31;
  const int wv    = tid >> 5;                         // 0..7
  const int m0    = blockIdx.y * 16;                  // blockIdx.y: 0..7
  const int n0    = (blockIdx.x * 8 + wv) * 32;       // blockIdx.x: 0..7
  const int laneM = lane & 15;
  const int laneN = lane & 15;
  const int koff  = (lane >> 4) << 1;                 // 0 or 2

  // Staging geometry: 256 threads x 8 iters x 16B = 32KB = 16 rows x 512 fl.
  const int srow0 = tid >> 7;                         // 0..1
  const int scol  = (tid & 127) << 2;                 // 0..508 step 4

  v8f acc0 = {};
  v8f acc1 = {};

  // ---- stage phase-1 A tile: rows of x ----
#pragma unroll
  for (int it = 0; it < 8; ++it) {
    const int r = srow0 + it * 2;
    async_copy_b128(lds_offset(&sA[r][scol]),
                    x + (m0 + r) * FEAT + scol);
  }
  wait_asynccnt0();
  __syncthreads();

  // Phase 1: K in [0,512) -> x @ kernel
  for (int k0 = 0; k0 < FEAT; k0 += 4) {
    const int k = k0 + koff;
    v2f a = *(const v2f*)&sA[laneM][k];               // ds_load_b64, no conflicts
    const float* b0 = kernelW + k * FOURU + n0 + laneN;
    const float* b1 = b0 + FOURU;
    v2f bb0; bb0.x = b0[0];  bb0.y = b1[0];
    v2f bb1; bb1.x = b0[16]; bb1.y = b1[16];
    acc0 = __builtin_amdgcn_wmma_f32_16x16x4_f32(false, a, false, bb0,
                                                 (short)0, acc0, false, false);
    acc1 = __builtin_amdgcn_wmma_f32_16x16x4_f32(false, a, false, bb1,
                                                 (short)0, acc1, false, false);
  }

  __syncthreads();  // all waves done reading phase-1 tile

  // ---- stage phase-2 A tile: gathered rows h_states[idx[m]] ----
#pragma unroll
  for (int it = 0; it < 8; ++it) {
    const int r = srow0 + it * 2;
    const int cls = idx[m0 + r];
    async_copy_b128(lds_offset(&sA[r][scol]),
                    h_states + cls * U + scol);
  }
  wait_asynccnt0();
  __syncthreads();

  // Phase 2: K in [512,1024) -> h_states[idx] @ rec_kernel
  for (int k0 = 0; k0 < U; k0 += 4) {
    const int k = k0 + koff;
    v2f a = *(const v2f*)&sA[laneM][k];
    const float* b0 = recW + k * FOURU + n0 + laneN;
    const float* b1 = b0 + FOURU;
    v2f bb0; bb0.x = b0[0];  bb0.y = b1[0];
    v2f bb1; bb1.x = b0[16]; bb1.y = b1[16];
    acc0 = __builtin_amdgcn_wmma_f32_16x16x4_f32(false, a, false, bb0,
                                                 (short)0, acc0, false, false);
    acc1 = __builtin_amdgcn_wmma_f32_16x16x4_f32(false, a, false, bb1,
                                                 (short)0, acc1, false, false);
  }

  // C/D layout (ISA 7.12.2): VGPR r -> lanes 0-15: M=r, lanes 16-31: M=r+8.
  const int row_base = m0 + ((lane >> 4) << 3);
  const int ncol     = n0 + laneN;
#pragma unroll
  for (int r = 0; r < 8; ++r) {
    const int row = row_base + r;
    z[row * FOURU + ncol +  0] = acc0[r];
    z[row * FOURU + ncol + 16] = acc1[r];
  }
}

// ---------------------------------------------------------------------------
// Step 3: gates + cell update + scatter-max into new_h/new_c.
// Float atomic max via sign-aware integer atomics (native global_atomic_max_i32
// / global_atomic_min_u32 — monotone bit encoding of IEEE754).
// ---------------------------------------------------------------------------
__device__ __forceinline__ void atomicMaxFloat(float* addr, float v) {
  if (v >= 0.0f) {
    atomicMax((int*)addr, __float_as_int(v));
  } else {
    atomicMin((unsigned int*)addr, __float_as_uint(v));
  }
}

__device__ __forceinline__ float sigmoidf(float v) {
  return 1.0f / (1.0f + expf(-v));
}

__global__ __launch_bounds__(256) void finalize_kernel(
    const float* __restrict__ z,         // (128,2048) raw
    const float* __restrict__ bias,      // (2048)
    const float* __restrict__ c_states,  // (1024,512)
    const int*   __restrict__ idx,       // (128)
    float* __restrict__ h_out,           // (128,512)
    float* __restrict__ new_h,           // (1024,512) pre-seeded with h_states
    float* __restrict__ new_c)           // (1024,512) pre-seeded with c_states
{
  const int gid = blockIdx.x * 256 + threadIdx.x;     // 0..65535
  const int b   = gid >> 9;
  const int u   = gid & (U - 1);
  const int cls = idx[b];

  const float* zr = z + b * FOURU + u;
  const float zi = zr[0 * U] + bias[0 * U + u];
  const float zf = zr[1 * U] + bias[1 * U + u];
  const float zg = zr[2 * U] + bias[2 * U + u];
  const float zo = zr[3 * U] + bias[3 * U + u];

  const float ig = sigmoidf(zi);
  const float fg = sigmoidf(zf);
  const float gg = tanhf(zg);
  const float og = sigmoidf(zo);

  const float c0 = c_states[cls * U + u];
  const float c  = fg * c0 + ig * gg;
  const float h  = og * tanhf(c);

  h_out[b * U + u] = h;
  atomicMaxFloat(new_h + cls * U + u, h);
  atomicMaxFloat(new_c + cls * U + u, c);
}

// ---------------------------------------------------------------------------
// Launch
// ---------------------------------------------------------------------------
extern "C" void kernel_launch(void* const* d_in, const int* in_sizes, int n_in,
                              void* d_out, int out_size, void* d_ws, size_t ws_size,
                              hipStream_t stream) {
  const float* x        = (const float*)d_in[0];  // (128,512)
  const float* logits   = (const float*)d_in[1];  // (128,1024)
  const float* h_states = (const float*)d_in[2];  // (1024,512)
  const float* c_states = (const float*)d_in[3];  // (1024,512)
  const float* kernelW  = (const float*)d_in[4];  // (512,2048)
  const float* recW     = (const float*)d_in[5];  // (512,2048)
  const float* bias     = (const float*)d_in[6];  // (2048)

  float* out   = (float*)d_out;
  float* h_out = out;                     // (128,512)
  float* new_h = out + BS * U;            // (1024,512)
  float* new_c = new_h + NC * U;          // (1024,512)

  int*   idx = (int*)d_ws;                                  // 128 ints
  float* z   = (float*)((char*)d_ws + 1024);                // (128,2048) = 1MB

  // Seed scatter outputs with the old states (stream-ordered before finalize).
  hipMemcpyAsync(new_h, h_states, (size_t)NC * U * sizeof(float),
                 hipMemcpyDeviceToDevice, stream);
  hipMemcpyAsync(new_c, c_states, (size_t)NC * U * sizeof(float),
                 hipMemcpyDeviceToDevice, stream);

  argmax_kernel<<<BS, 256, 0, stream>>>(logits, idx);
  gemm_z_kernel<<<dim3(8, 8), 256, 0, stream>>>(x, h_states, kernelW, recW,
                                                idx, z);
  finalize_kernel<<<(BS * U) / 256, 256, 0, stream>>>(z, bias, c_states, idx,
                                                      h_out, new_h, new_c);
}